// stressUpdateLayer_49392123904158
// MI455X (gfx1250) — compile-verified
//
#include <hip/hip_runtime.h>
#include <stdint.h>

// Problem constants (from reference): B=128, T=2048, NM=8
#define BSZ   128
#define TT    2048
#define NMODE 8
#define NG    9          // NM+1 g-values; also 9 tensor elements (3x3)
#define NCH   32         // chunks along T
#define LCH   64         // TT / NCH

typedef uint32_t u32x4 __attribute__((ext_vector_type(4)));
typedef uint32_t u32x8 __attribute__((ext_vector_type(8)));

// --- CDNA5 data-mover helpers (gfx1250) --------------------------------------

// Low 32 bits of a flat (generic) LDS address are the wave-relative LDS byte
// offset (aperture layout: generic = {SHARED_BASE_HI, offset32}).
__device__ __forceinline__ uint32_t lds_off_of(const void* p) {
  return (uint32_t)(uintptr_t)p;
}

// Tensor Data Mover: 1D contiguous copy of `ndw` dwords global -> LDS.
// D# built per CDNA5 ISA (2-group form, VADDR2/3 = NULL):
//   group0: count=1 | lds_addr | global_addr[56:0] | type=2
//   group1: data_size=4B, tensor_dim0=ndw, tensor_dim1=1,
//           tile_dim0=ndw (<=65535), tile_dim1=1, stride0=ndw
// Wave-level op (EXEC ignored), tracked by TENSORcnt.
__device__ __forceinline__ void tdm_load_1d(uint32_t lds_off,
                                            const void* gsrc, uint32_t ndw) {
  const uint64_t ga = (uint64_t)(uintptr_t)gsrc;
  u32x4 g0;
  g0[0] = 1u;                                              // count=1, flags=0
  g0[1] = lds_off;                                         // lds_addr (bytes)
  g0[2] = (uint32_t)ga;                                    // global_addr[31:0]
  g0[3] = ((uint32_t)(ga >> 32) & 0x01FFFFFFu)             // global_addr[56:32]
          | (2u << 30);                                    // type=2 ("image")
  u32x8 g1;
  g1[0] = (2u << 16);                                      // data_size=2 -> 4B
  g1[1] = (ndw & 0xFFFFu) << 16;                           // tensor_dim0[15:0]
  g1[2] = (ndw >> 16) | (1u << 16);                        // dim0[31:16], tensor_dim1=1
  g1[3] = (ndw & 0xFFFFu) << 16;                           // tensor_dim1[31:16]=0, tile_dim0=ndw
  g1[4] = 1u;                                              // tile_dim1=1, tile_dim2=0
  g1[5] = ndw;                                             // tensor_dim0_stride[31:0]
  g1[6] = 0u;                                              // stride0[47:32], stride1[15:0]
  g1[7] = 0u;                                              // stride1[47:16]
  asm volatile("tensor_load_to_lds %0, %1"
               :: "s"(g0), "s"(g1)
               : "memory");
}

// Per-lane async copy of n16 * 16 bytes, 16B granules, striped across wave32.
// global_load_async_to_lds_b128: VDST = LDS byte address, VADDR = 64-bit global.
__device__ __forceinline__ void async_copy16(uint32_t lds_off,
                                             const float* __restrict__ gsrc,
                                             int n16, int lane) {
  uint64_t gb = (uint64_t)(uintptr_t)gsrc;
  for (int u = lane; u < n16; u += 32) {
    uint32_t lo = lds_off + (uint32_t)u * 16u;
    uint64_t ga = gb + (uint64_t)u * 16u;
    asm volatile("global_load_async_to_lds_b128 %0, %1, off"
                 :: "v"(lo), "v"(ga)
                 : "memory");
  }
}

__device__ __forceinline__ void wait_async0() {
  asm volatile("s_wait_asynccnt 0" ::: "memory");
}

// --- chunked scan kernel (pass 1: FINAL=false, pass 3: FINAL=true) -----------
//
// Block = one (batch, chunk) pair, 1 wave (32 lanes).
// TDM stages the S_e / PronyParams chunk regions; async-LDS loads stage the
// t vector. Lanes 0..8 then each own one 3x3 element and carry all 8 mode
// recurrences in registers (no cross-lane reductions needed).

template <bool FINAL>
__global__ void __launch_bounds__(32)
chunk_scan(const float* __restrict__ S_e,    // [B][T][9]
           const float* __restrict__ tvec,   // [B][T]
           const float* __restrict__ P,      // [B][T][17]  (tau[8], g[9])
           float* __restrict__ Aprod,        // [B][NCH][NM]        (pass1 out)
           float* __restrict__ Qend,         // [B][NCH][NM][9]     (pass1 out)
           const float* __restrict__ Q0,     // [B][NCH][NM][9]     (pass3 in)
           float* __restrict__ outS,         // [B][T][9]
           float* __restrict__ outSinf,      // [B][T][9]
           float* __restrict__ outQ)         // [B][T][9]
{
  __shared__ __align__(16) float smem[LCH * 9 + LCH + LCH * 17]; // 1728 floats
  float* smS = smem;               // [LCH][9]
  float* smT = smem + LCH * 9;     // [LCH]
  float* smP = smem + LCH * 10;    // [LCH][17]

  const int lane = threadIdx.x;
  const int b = blockIdx.x / NCH;
  const int c = blockIdx.x % NCH;
  const int t0 = c * LCH;

  const float* gS = S_e  + ((size_t)b * TT + t0) * 9;
  const float* gT = tvec +  (size_t)b * TT + t0;
  const float* gP = P    + ((size_t)b * TT + t0) * 17;

  // Stage chunk inputs into LDS: TDM for the two big contiguous regions,
  // per-lane async-LDS loads for the small t region.
  tdm_load_1d(lds_off_of(smS), gS, LCH * 9);      // 576 dwords
  tdm_load_1d(lds_off_of(smP), gP, LCH * 17);     // 1088 dwords
  async_copy16(lds_off_of(smT), gT, (LCH * 4) / 16, lane);  // 16 granules
  wait_async0();
  __builtin_amdgcn_s_wait_tensorcnt(0);           // single wave: waits suffice

  if (lane >= NG) return;
  const int e = lane;

  // ---- carried state entering this chunk ----
  float tau_p[NMODE], g_p[NMODE];
  float t_p, S_p;
  if (c == 0) {
    // reference init: t_p = -t[:,1]; S_p = 0; tau_p = tau[:,0]; g_p = g_i[:,0]
    t_p = -tvec[(size_t)b * TT + 1];
    S_p = 0.0f;
    const float* Pr = P + (size_t)b * TT * 17;   // row 0
    float gs = 0.0f;
#pragma unroll
    for (int j = 0; j < NG; ++j) gs += Pr[8 + j];
    float inv = __builtin_amdgcn_rcpf(gs);
#pragma unroll
    for (int m = 0; m < NMODE; ++m) { tau_p[m] = Pr[m]; g_p[m] = Pr[9 + m] * inv; }
  } else {
    const int r = t0 - 1;
    t_p = tvec[(size_t)b * TT + r];
    S_p = S_e[((size_t)b * TT + r) * 9 + e];
    const float* Pr = P + ((size_t)b * TT + r) * 17;
    float gs = 0.0f;
#pragma unroll
    for (int j = 0; j < NG; ++j) gs += Pr[8 + j];
    float inv = __builtin_amdgcn_rcpf(gs);
#pragma unroll
    for (int m = 0; m < NMODE; ++m) { tau_p[m] = Pr[m]; g_p[m] = Pr[9 + m] * inv; }
  }

  float Q[NMODE], Aacc[NMODE];
  const size_t cbase = ((size_t)b * NCH + c) * NMODE;
#pragma unroll
  for (int m = 0; m < NMODE; ++m) {
    Q[m]    = FINAL ? Q0[(cbase + m) * NG + e] : 0.0f;
    Aacc[m] = 1.0f;
  }

  // ---- in-chunk sequential scan (LCH = 64 steps) ----
  for (int k = 0; k < LCH; ++k) {
    const float  t_c = smT[k];
    const float* Pr  = smP + k * 17;
    float gs = 0.0f;
#pragma unroll
    for (int j = 0; j < NG; ++j) gs += Pr[8 + j];
    const float inv = __builtin_amdgcn_rcpf(gs);
    const float S_c = smS[k * 9 + e];
    const float dt  = t_c - t_p;
    const float dS  = S_c - S_p;
    float qsum = 0.0f;
#pragma unroll
    for (int m = 0; m < NMODE; ++m) {
      const float tau_c   = Pr[m];
      const float g_c     = Pr[9 + m] * inv;
      const float g_bar   = 0.5f * (g_c + g_p[m]);
      const float tau_bar = 0.5f * (tau_c + tau_p[m]);
      const float ex = __expf(-0.5f * dt * __builtin_amdgcn_rcpf(tau_bar));
      const float a  = ex * ex;
      Q[m] = a * Q[m] + (ex * g_bar) * dS;
      if (!FINAL) Aacc[m] *= a;
      if (FINAL)  qsum += Q[m];
      tau_p[m] = tau_c;
      g_p[m]   = g_c;
    }
    if (FINAL) {
      const float ginf = Pr[8] * inv;
      const float Sinf = ginf * S_c;
      const size_t o = ((size_t)b * TT + t0 + k) * 9 + e;
      outS[o]    = Sinf + qsum;
      outSinf[o] = Sinf;
      outQ[o]    = qsum;
    }
    t_p = t_c;
    S_p = S_c;
  }

  if (!FINAL) {
#pragma unroll
    for (int m = 0; m < NMODE; ++m) Qend[(cbase + m) * NG + e] = Q[m];
    if (e == 0) {
#pragma unroll
      for (int m = 0; m < NMODE; ++m) Aprod[cbase + m] = Aacc[m];
    }
  }
}

// --- pass 2: chunk-level exclusive prefix per (b, mode, elem) ----------------
__global__ void __launch_bounds__(256)
chunk_prefix(const float* __restrict__ Aprod,   // [B][NCH][NM]
             const float* __restrict__ Qend,    // [B][NCH][NM][9]
             float* __restrict__ Q0)            // [B][NCH][NM][9]
{
  const int idx = blockIdx.x * blockDim.x + threadIdx.x;
  if (idx >= BSZ * NMODE * NG) return;
  const int e = idx % NG;
  const int m = (idx / NG) % NMODE;
  const int b = idx / (NG * NMODE);
  float q = 0.0f;
  for (int c = 0; c < NCH; ++c) {
    const size_t base = ((size_t)b * NCH + c) * NMODE + m;
    Q0[base * NG + e] = q;
    q = Aprod[base] * q + Qend[base * NG + e];
  }
}

// --- launcher ----------------------------------------------------------------
extern "C" void kernel_launch(void* const* d_in, const int* in_sizes, int n_in,
                              void* d_out, int out_size, void* d_ws, size_t ws_size,
                              hipStream_t stream) {
  const float* S_e = (const float*)d_in[0];   // [B,T,3,3]
  const float* t   = (const float*)d_in[1];   // [B,T]
  const float* P   = (const float*)d_in[2];   // [B,T,17]

  // Workspace: Aprod (32K floats) | Qend (288K) | Q0 (288K) ~= 2.5 MB
  float* Aprod = (float*)d_ws;
  float* Qend  = Aprod + (size_t)BSZ * NCH * NMODE;
  float* Q0    = Qend  + (size_t)BSZ * NCH * NMODE * NG;

  float* outS    = (float*)d_out;
  float* outSinf = outS    + (size_t)BSZ * TT * 9;
  float* outQ    = outSinf + (size_t)BSZ * TT * 9;

  dim3 grid(BSZ * NCH);   // 4096 blocks, 1 wave each

  chunk_scan<false><<<grid, 32, 0, stream>>>(S_e, t, P, Aprod, Qend,
                                             nullptr, nullptr, nullptr, nullptr);

  const int nth = BSZ * NMODE * NG;  // 9216
  chunk_prefix<<<(nth + 255) / 256, 256, 0, stream>>>(Aprod, Qend, Q0);

  chunk_scan<true><<<grid, 32, 0, stream>>>(S_e, t, P, nullptr, nullptr, Q0,
                                            outS, outSinf, outQ);
}